// SurrogateKAN_26568667693277
// MI455X (gfx1250) — compile-verified
//
#include <hip/hip_runtime.h>

// ---------------------------------------------------------------------------
// KAN forward on MI455X (gfx1250): spline einsum + linear + bias fused into a
// single fp32 WMMA GEMM per layer (V_WMMA_F32_16X16X4_F32), full reference
// precision. A-matrix fragments are generated in-register (RBF exps), B
// matrices are pre-packed into the WMMA lane layout by a prep kernel.
// ---------------------------------------------------------------------------

typedef __attribute__((ext_vector_type(2))) float v2f;
typedef __attribute__((ext_vector_type(8))) float v8f;

#define GRIDN 12
#define INV_BW (11.0f / 6.0f)   // 1 / bandwidth, bandwidth = 6/11
#define CSTEP  (6.0f / 11.0f)   // center spacing, centers = -3 + g*CSTEP

// Layer K totals (basis + linear + bias, padded to multiple of 4):
//  L1: 9*12 + 9 + 1 = 118 -> 120  (30 K-steps), N: 24 -> 2 tiles of 16
//  L2: 24*12 + 24 + 1 = 313 -> 316 (79 K-steps), N: 12 -> 1 tile
//  L3: 12*12 + 12 + 1 = 157 -> 160 (40 K-steps), N: 3  -> 1 tile
#define NSTEP1 30
#define NSTEP2 79
#define NSTEP3 40
#define NT1 2
#define NT2 1
#define NT3 1
#define B1_FLOATS (NSTEP1 * NT1 * 64)           // 3840
#define B2_FLOATS (NSTEP2 * NT2 * 64)           // 5056
#define B3_FLOATS (NSTEP3 * NT3 * 64)           // 2560
#define B1_OFF 0
#define B2_OFF (B1_FLOATS)
#define B3_OFF (B1_FLOATS + B2_FLOATS)
#define BTOTAL (B1_FLOATS + B2_FLOATS + B3_FLOATS)  // 11456 floats (~45 KB)

// ---------------------------------------------------------------------------
// Prep kernel: pack B matrices into d_ws in WMMA-B fragment order.
// Element e of a layer maps to: lane = (e/2)%32, v = e%2, q = (e/2)/32,
// t = q/NT, nt = q%NT; then k = 4t + 2*(lane/16) + v, n = nt*16 + lane%16.
// ---------------------------------------------------------------------------
__global__ __launch_bounds__(256) void kan_pack(
    const float* __restrict__ c1, const float* __restrict__ w1, const float* __restrict__ b1,
    const float* __restrict__ c2, const float* __restrict__ w2, const float* __restrict__ b2,
    const float* __restrict__ c3, const float* __restrict__ w3, const float* __restrict__ b3,
    const float* __restrict__ base, float* __restrict__ ws) {
  int idx = blockIdx.x * blockDim.x + threadIdx.x;
  if (idx >= BTOTAL) return;

  int layer, e, nt_count;
  if (idx < B2_OFF)       { layer = 0; e = idx;           nt_count = NT1; }
  else if (idx < B3_OFF)  { layer = 1; e = idx - B2_OFF;  nt_count = NT2; }
  else                    { layer = 2; e = idx - B3_OFF;  nt_count = NT3; }

  int v    = e & 1;
  int lane = (e >> 1) & 31;
  int q    = (e >> 1) >> 5;
  int t    = q / nt_count;
  int nt   = q - t * nt_count;
  int k    = 4 * t + 2 * (lane >> 4) + v;
  int n    = nt * 16 + (lane & 15);

  float val = 0.0f;
  if (layer == 0) {                      // I=9, O=24, KB=108
    if (n < 24) {
      if (k < 108)      { int i = k / 12, g = k - i * 12; val = c1[(i * 24 + n) * 12 + g]; }
      else if (k < 117) { val = w1[n * 9 + (k - 108)]; }
      else if (k == 117){ val = b1[n]; }
    }
  } else if (layer == 1) {               // I=24, O=12, KB=288
    if (n < 12) {
      if (k < 288)      { int i = k / 12, g = k - i * 12; val = c2[(i * 12 + n) * 12 + g]; }
      else if (k < 312) { val = w2[n * 24 + (k - 288)]; }
      else if (k == 312){ val = b2[n]; }
    }
  } else {                               // I=12, O=3, KB=144 (base folded in)
    if (n < 3) {
      if (k < 144)      { int i = k / 12, g = k - i * 12; val = c3[(i * 3 + n) * 12 + g]; }
      else if (k < 156) { val = w3[n * 12 + (k - 144)]; }
      else if (k == 156){ val = b3[n] + base[n]; }
    }
  }
  ws[idx] = val;
}

// ---------------------------------------------------------------------------
// A-fragment element: k < I*G -> RBF basis of a[i]; next I -> linear passthru;
// one bias column of 1.0; zero pad.
// ---------------------------------------------------------------------------
template <int I>
__device__ __forceinline__ float afrag_elem(const float* __restrict__ a, int k) {
  const int KB = I * GRIDN;
  float r;
  if (k < KB) {
    int i = (int)((unsigned)k / 12u);
    int g = k - i * 12;
    float d = (a[i] - (-3.0f + CSTEP * (float)g)) * INV_BW;
    r = __expf(-0.5f * d * d);
  } else if (k < KB + I) {
    r = a[k - KB];
  } else if (k == KB + I) {
    r = 1.0f;
  } else {
    r = 0.0f;
  }
  return r;
}

// One fused KAN layer: C(16xN) += A(16xK) * B(KxN) via v_wmma_f32_16x16x4_f32.
template <int I, int NSTEPS, int NT>
__device__ __forceinline__ void kan_layer_wmma(const float* __restrict__ arow,
                                               const float* __restrict__ Bsh,
                                               int lane, int half,
                                               v8f& c0, v8f& c1) {
#pragma unroll 4
  for (int t = 0; t < NSTEPS; ++t) {
    int k0 = 4 * t + 2 * half;
    v2f a;
    a.x = afrag_elem<I>(arow, k0);
    a.y = afrag_elem<I>(arow, k0 + 1);
    v2f b0 = *(const v2f*)(Bsh + (t * NT + 0) * 64 + lane * 2);
    c0 = __builtin_amdgcn_wmma_f32_16x16x4_f32(false, a, false, b0, (short)0, c0,
                                               false, false);
    if (NT == 2) {
      v2f b1 = *(const v2f*)(Bsh + (t * NT + 1) * 64 + lane * 2);
      c1 = __builtin_amdgcn_wmma_f32_16x16x4_f32(false, a, false, b1, (short)0, c1,
                                                 false, false);
    }
  }
}

__device__ __forceinline__ float silu_f(float z) {
  return z * (1.0f / (1.0f + __expf(-z)));
}

// ---------------------------------------------------------------------------
// Main kernel: 8 waves/block, each wave owns a 16-row batch tile per
// iteration; 4 iterations per block (512 rows) to amortize the B->LDS fill.
// Wave-private activation buffers in LDS give the C-tile -> A-row transpose.
// ---------------------------------------------------------------------------
__global__ __launch_bounds__(256, 2) void kan_forward(const float* __restrict__ x,
                                                      const float* __restrict__ ws,
                                                      float* __restrict__ out,
                                                      int tiles_per_wave) {
  __shared__ __align__(16) float Bsh[BTOTAL];
  __shared__ __align__(16) float act0[8][16][12];  // layer-0 input (9 used)
  __shared__ __align__(16) float act1[8][16][24];  // h1
  __shared__ __align__(16) float act2[8][16][12];  // h2
  __shared__ __align__(16) float cor[8][16][4];    // corr (3 used)

  const int tid = threadIdx.x;
  for (int i = tid; i < BTOTAL; i += 256) Bsh[i] = ws[i];
  __syncthreads();

  const int wv   = tid >> 5;
  const int lane = tid & 31;
  const int half = lane >> 4;   // 0: K even pair, 1: K odd pair (f32 WMMA A/B layout)
  const int m    = lane & 15;   // A-matrix row within tile
  const float* a0 = &act0[wv][m][0];
  const float* a1 = &act1[wv][m][0];
  const float* a2 = &act2[wv][m][0];

  for (int it = 0; it < tiles_per_wave; ++it) {
    const int rowbase = ((blockIdx.x * tiles_per_wave + it) * 8 + wv) * 16;

    // Stage this wave's 16x9 input tile into LDS (touch-once stream -> NT).
    for (int e = lane; e < 144; e += 32) {
      int r = e / 9, c = e - r * 9;
      act0[wv][r][c] = __builtin_nontemporal_load(&x[(size_t)(rowbase + r) * 9 + c]);
    }
    asm volatile("s_wait_dscnt 0" ::: "memory");

    // Layer 1: 16x120 @ 120x24 (+SiLU)
    v8f c0 = {}, c1 = {};
    kan_layer_wmma<9, NSTEP1, NT1>(a0, Bsh + B1_OFF, lane, half, c0, c1);
#pragma unroll
    for (int v = 0; v < 8; ++v) {
      int r = half * 8 + v;                 // C layout: lane holds rows 8*half+v, col m
      act1[wv][r][m] = silu_f(c0[v]);
      if (m < 8) act1[wv][r][16 + m] = silu_f(c1[v]);
    }
    asm volatile("s_wait_dscnt 0" ::: "memory");

    // Layer 2: 16x316 @ 316x12 (+SiLU)
    v8f d0 = {}, dz = {};
    kan_layer_wmma<24, NSTEP2, NT2>(a1, Bsh + B2_OFF, lane, half, d0, dz);
#pragma unroll
    for (int v = 0; v < 8; ++v) {
      int r = half * 8 + v;
      if (m < 12) act2[wv][r][m] = silu_f(d0[v]);
    }
    asm volatile("s_wait_dscnt 0" ::: "memory");

    // Layer 3: 16x160 @ 160x3 (bias+base folded)
    v8f e0 = {}, ez = {};
    kan_layer_wmma<12, NSTEP3, NT3>(a2, Bsh + B3_OFF, lane, half, e0, ez);
#pragma unroll
    for (int v = 0; v < 8; ++v) {
      int r = half * 8 + v;
      if (m < 3) cor[wv][r][m] = e0[v];
    }
    asm volatile("s_wait_dscnt 0" ::: "memory");

    // Softmax over 3 classes; lanes 0..15 each own one row.
    if (half == 0) {
      float p0 = cor[wv][m][0], p1 = cor[wv][m][1], p2 = cor[wv][m][2];
      float mx = fmaxf(p0, fmaxf(p1, p2));
      float q0 = __expf(p0 - mx), q1 = __expf(p1 - mx), q2 = __expf(p2 - mx);
      float inv = 1.0f / (q0 + q1 + q2);
      float* o = out + (size_t)(rowbase + m) * 3;
      __builtin_nontemporal_store(q0 * inv, o + 0);
      __builtin_nontemporal_store(q1 * inv, o + 1);
      __builtin_nontemporal_store(q2 * inv, o + 2);
    }
    asm volatile("s_wait_dscnt 0" ::: "memory");
  }
}

// ---------------------------------------------------------------------------
extern "C" void kernel_launch(void* const* d_in, const int* in_sizes, int n_in,
                              void* d_out, int out_size, void* d_ws, size_t ws_size,
                              hipStream_t stream) {
  const float* x    = (const float*)d_in[0];
  const float* c1   = (const float*)d_in[1];
  const float* w1   = (const float*)d_in[2];
  const float* b1   = (const float*)d_in[3];
  const float* c2   = (const float*)d_in[4];
  const float* w2   = (const float*)d_in[5];
  const float* b2   = (const float*)d_in[6];
  const float* c3   = (const float*)d_in[7];
  const float* w3   = (const float*)d_in[8];
  const float* b3   = (const float*)d_in[9];
  const float* base = (const float*)d_in[10];
  float* ws  = (float*)d_ws;
  float* out = (float*)d_out;

  const int Btot = in_sizes[0] / 9;        // 524288
  const int TILES = 4;                     // 16-row tiles per wave
  const int blocks = Btot / (128 * TILES); // 1024 blocks of 256 threads

  kan_pack<<<(BTOTAL + 255) / 256, 256, 0, stream>>>(c1, w1, b1, c2, w2, b2,
                                                     c3, w3, b3, base, ws);
  kan_forward<<<blocks, 256, 0, stream>>>(x, ws, out, TILES);
}